// HyperbolicUnionRGCNLayer_26680336843172
// MI455X (gfx1250) — compile-verified
//
#include <hip/hip_runtime.h>
#include <hip/hip_bf16.h>

typedef __attribute__((ext_vector_type(2))) float v2f;
typedef __attribute__((ext_vector_type(8))) float v8f;

#define DFEAT 128
#define SQRT_C 0.1f      // sqrt(0.01)
#define CLIPV 10.0f

// ---------------------------------------------------------------------------
// 0) zero agg (d_out) and indeg
// ---------------------------------------------------------------------------
__global__ void __launch_bounds__(256)
init_zero_kernel(float* __restrict__ agg, int* __restrict__ indeg,
                 int nAgg, int nDeg) {
    int i = blockIdx.x * blockDim.x + threadIdx.x;
    int stride = gridDim.x * blockDim.x;
    for (int j = i; j < nAgg; j += stride) agg[j] = 0.0f;
    for (int j = i; j < nDeg; j += stride) indeg[j] = 0;
}

// ---------------------------------------------------------------------------
// 1) per-node log map + radius.  One wave32 per node, 4 floats per lane.
// ---------------------------------------------------------------------------
__global__ void __launch_bounds__(256)
node_logmap_kernel(const float* __restrict__ x, float* __restrict__ ht,
                   float* __restrict__ rad, int Nn) {
    const int wave = threadIdx.x >> 5;
    const int lane = threadIdx.x & 31;
    const int node = blockIdx.x * 8 + wave;
    if (node >= Nn) return;

    const float4 v = ((const float4*)(x + (size_t)node * DFEAT))[lane];
    float ss = v.x * v.x + v.y * v.y + v.z * v.z + v.w * v.w;
    #pragma unroll
    for (int m = 16; m > 0; m >>= 1) ss += __shfl_xor(ss, m, 32);

    float n   = fmaxf(sqrtf(ss), 1e-10f);
    float scn = SQRT_C * n;
    // artanh with reference clipping
    float xa  = fminf(fmaxf(scn, -1.0f + 1e-7f), 1.0f - 1e-7f);
    float art = 0.5f * logf((1.0f + xa) / (1.0f - xa));
    float scl = art / scn;

    float4 o;
    o.x = v.x * scl; o.y = v.y * scl; o.z = v.z * scl; o.w = v.w * scl;
    ((float4*)(ht + (size_t)node * DFEAT))[lane] = o;

    if (lane == 0) {
        float rr = fminf(scn, 1.0f - 1e-5f);
        rad[node] = (2.0f / SQRT_C) * 0.5f * logf((1.0f + rr) / (1.0f - rr));
    }
}

// ---------------------------------------------------------------------------
// 2) fp32 WMMA GEMM:  C[M x 128] = A[M x 128] @ B[128 x 128]
//    One wave computes a 16-row x 128-col strip with V_WMMA_F32_16X16X4_F32.
//    B is staged in LDS (64 KB).  8 waves / block -> 128 rows per block.
// ---------------------------------------------------------------------------
__global__ void __launch_bounds__(256)
gemm_wmma_f32_kernel(const float* __restrict__ A, const float* __restrict__ B,
                     float* __restrict__ C, int M) {
    __shared__ float Blds[DFEAT * DFEAT];

    const int tid = threadIdx.x;
    {   // cooperative 64KB B load, float4
        const float4* b4 = (const float4*)B;
        float4* l4 = (float4*)Blds;
        for (int i = tid; i < DFEAT * DFEAT / 4; i += 256) l4[i] = b4[i];
    }
    __syncthreads();

    const int wave    = tid >> 5;
    const int lane    = tid & 31;
    const int strip   = blockIdx.x * 8 + wave;       // 16-row strip
    const int nstrips = (M + 15) >> 4;
    if (strip >= nstrips) return;                    // wave-uniform: EXEC stays full

    const int rhalf = lane >> 4;                     // 0 -> k,k+1 ; 1 -> k+2,k+3
    const int l15   = lane & 15;
    int row = strip * 16 + l15;
    if (row >= M) row = M - 1;                       // clamp loads, guard stores

    v8f acc[8] = {};                                 // 8 col tiles of 16

    const float* arow = A + (size_t)row * DFEAT;
    #pragma unroll 4
    for (int k = 0; k < DFEAT; k += 4) {
        v2f a;
        const float* ap = arow + k + rhalf * 2;
        a.x = ap[0];
        a.y = ap[1];
        const float* bp = Blds + (k + rhalf * 2) * DFEAT + l15;
        #pragma unroll
        for (int t = 0; t < 8; ++t) {
            v2f b;
            b.x = bp[t * 16];
            b.y = bp[t * 16 + DFEAT];
            acc[t] = __builtin_amdgcn_wmma_f32_16x16x4_f32(
                false, a, false, b, (short)0, acc[t], false, false);
        }
    }

    // C/D layout: vgpr j, lanes 0-15 -> M=j, lanes 16-31 -> M=8+j
    const int rbase = strip * 16 + rhalf * 8;
    if (strip * 16 + 15 < M) {
        #pragma unroll
        for (int t = 0; t < 8; ++t) {
            const int col = t * 16 + l15;
            #pragma unroll
            for (int j = 0; j < 8; ++j)
                C[(size_t)(rbase + j) * DFEAT + col] = acc[t][j];
        }
    } else {
        for (int t = 0; t < 8; ++t) {
            const int col = t * 16 + l15;
            for (int j = 0; j < 8; ++j) {
                const int r = rbase + j;
                if (r < M) C[(size_t)r * DFEAT + col] = acc[t][j];
            }
        }
    }
}

// ---------------------------------------------------------------------------
// 3) in-degree
// ---------------------------------------------------------------------------
__global__ void __launch_bounds__(256)
indeg_kernel(const int* __restrict__ dst, int* __restrict__ indeg, int Ne) {
    int e = blockIdx.x * blockDim.x + threadIdx.x;
    if (e < Ne) atomicAdd(&indeg[dst[e]], 1);
}

// ---------------------------------------------------------------------------
// 4) per-edge message + scatter-add.  One wave32 per edge, float4 per lane.
// ---------------------------------------------------------------------------
__global__ void __launch_bounds__(256)
edge_scatter_kernel(const float* __restrict__ Hn, const float* __restrict__ RW,
                    const float* __restrict__ rad, const int* __restrict__ src,
                    const int* __restrict__ dst, const int* __restrict__ et,
                    float* __restrict__ agg, int Ne) {
    const int wave = threadIdx.x >> 5;
    const int lane = threadIdx.x & 31;
    const int e = blockIdx.x * 8 + wave;
    if (e >= Ne) return;

    const int s = src[e];
    const int d = dst[e];
    const int r = et[e];
    const float w = expf(-fabsf(rad[s] - rad[d]));

    const float4 hv = ((const float4*)(Hn + (size_t)s * DFEAT))[lane];
    const float4 rv = ((const float4*)(RW + (size_t)r * DFEAT))[lane];
    float* out = agg + (size_t)d * DFEAT + lane * 4;
    atomicAdd(out + 0, (hv.x + rv.x) * w);
    atomicAdd(out + 1, (hv.y + rv.y) * w);
    atomicAdd(out + 2, (hv.z + rv.z) * w);
    atomicAdd(out + 3, (hv.w + rv.w) * w);
}

// ---------------------------------------------------------------------------
// 5) finalize: clip(agg*norm) + self-loop, clip, exp-map.  Wave per node.
//    agg aliases out (d_out); each thread RMWs only its own float4.
// ---------------------------------------------------------------------------
__global__ void __launch_bounds__(256)
finalize_kernel(const float* __restrict__ norm, const int* __restrict__ indeg,
                const float* __restrict__ Hl, const float* __restrict__ He,
                float* __restrict__ out, int Nn) {
    const int wave = threadIdx.x >> 5;
    const int lane = threadIdx.x & 31;
    const int node = blockIdx.x * 8 + wave;
    if (node >= Nn) return;

    const float nm = norm[node];
    const float* lrow = (indeg[node] > 0 ? Hl : He) + (size_t)node * DFEAT;
    const float4 a = ((const float4*)(out + (size_t)node * DFEAT))[lane];
    const float4 l = ((const float4*)lrow)[lane];

    float4 h;
    h.x = fminf(fmaxf(fminf(fmaxf(a.x * nm, -CLIPV), CLIPV) + l.x, -CLIPV), CLIPV);
    h.y = fminf(fmaxf(fminf(fmaxf(a.y * nm, -CLIPV), CLIPV) + l.y, -CLIPV), CLIPV);
    h.z = fminf(fmaxf(fminf(fmaxf(a.z * nm, -CLIPV), CLIPV) + l.z, -CLIPV), CLIPV);
    h.w = fminf(fmaxf(fminf(fmaxf(a.w * nm, -CLIPV), CLIPV) + l.w, -CLIPV), CLIPV);

    float ss = h.x * h.x + h.y * h.y + h.z * h.z + h.w * h.w;
    #pragma unroll
    for (int m = 16; m > 0; m >>= 1) ss += __shfl_xor(ss, m, 32);

    float n   = fmaxf(sqrtf(ss), 1e-10f);
    float scn = SQRT_C * n;
    float scl = tanhf(scn) / scn;

    float4 o;
    o.x = h.x * scl; o.y = h.y * scl; o.z = h.z * scl; o.w = h.w * scl;
    ((float4*)(out + (size_t)node * DFEAT))[lane] = o;
}

// ---------------------------------------------------------------------------
extern "C" void kernel_launch(void* const* d_in, const int* in_sizes, int n_in,
                              void* d_out, int out_size, void* d_ws, size_t ws_size,
                              hipStream_t stream) {
    const float* h_hyper = (const float*)d_in[0];
    const float* rel_emb = (const float*)d_in[1];
    const float* normv   = (const float*)d_in[2];
    const int*   src     = (const int*)d_in[3];
    const int*   dst     = (const int*)d_in[4];
    const int*   etype   = (const int*)d_in[5];
    const float* Wn      = (const float*)d_in[6];
    const float* Wl      = (const float*)d_in[7];
    const float* We      = (const float*)d_in[8];
    float* out = (float*)d_out;

    const int N = in_sizes[0] / DFEAT;
    const int R = in_sizes[1] / DFEAT;
    const int E = in_sizes[3];

    // workspace carve-up (floats)
    const size_t ND = (size_t)N * DFEAT;
    float* ht  = (float*)d_ws;
    float* Hn  = ht + ND;
    float* Hl  = Hn + ND;
    float* He  = Hl + ND;
    float* RW  = He + ND;
    float* rad = RW + (size_t)R * DFEAT;
    int*  indeg = (int*)(rad + N);

    const int nodeBlocks = (N + 7) / 8;   // wave per node
    const int edgeBlocks = (E + 7) / 8;   // wave per edge

    // 0) zero agg (d_out) + indeg
    init_zero_kernel<<<2048, 256, 0, stream>>>(out, indeg, (int)ND, N);

    // 1) log map + radius
    node_logmap_kernel<<<nodeBlocks, 256, 0, stream>>>(h_hyper, ht, rad, N);

    // 2) GEMMs (WMMA fp32): Hn, Hl, He over nodes; RW over relations
    {
        const int gN = (((N + 15) >> 4) + 7) / 8;
        gemm_wmma_f32_kernel<<<gN, 256, 0, stream>>>(ht, Wn, Hn, N);
        gemm_wmma_f32_kernel<<<gN, 256, 0, stream>>>(ht, Wl, Hl, N);
        gemm_wmma_f32_kernel<<<gN, 256, 0, stream>>>(ht, We, He, N);
        const int gR = (((R + 15) >> 4) + 7) / 8;
        gemm_wmma_f32_kernel<<<gR, 256, 0, stream>>>(rel_emb, Wn, RW, R);
    }

    // 3) in-degree
    indeg_kernel<<<(E + 255) / 256, 256, 0, stream>>>(dst, indeg, E);

    // 4) edge messages + scatter
    edge_scatter_kernel<<<edgeBlocks, 256, 0, stream>>>(Hn, RW, rad, src, dst,
                                                        etype, out, E);

    // 5) finalize + exp map
    finalize_kernel<<<nodeBlocks, 256, 0, stream>>>(normv, indeg, Hl, He, out, N);
}